// LinearGaussianEnergyFunction_35905926594842
// MI455X (gfx1250) — compile-verified
//
#include <hip/hip_runtime.h>
#include <math.h>

// LinearGaussianEnergy for MI455X (gfx1250, wave32).
// quad = v^T Pc^-1 v via blocked unpivoted LU solve (NB=128); the O(D^3)
// trailing updates run on V_WMMA_F32_16X16X4_F32 with k-pair-interleaved LDS
// tiles (fragments load as single ds_load_b64 into even VGPR pairs) and a
// double-buffered async global->LDS pipeline (GLOBAL_LOAD_ASYNC_TO_LDS_B32,
// ASYNCcnt-tracked; register-staged fallback if the builtin is absent).
// LU is factored in-place in d_out (64MB) and fully overwritten by the final
// elementwise pass; d_ws holds only the solve vector + quad scalar.

#define Dn   4096
#define NB   128
#define EPSF 1e-6f
#define TWOPI_F 6.28318530717958647692f

typedef float v2f __attribute__((ext_vector_type(2)));
typedef float v8f __attribute__((ext_vector_type(8)));

#if defined(__AMDGCN__) && __has_builtin(__builtin_amdgcn_global_load_async_to_lds_b32)
#define USE_ASYNC_LDS 1
#else
#define USE_ASYNC_LDS 0
#endif

#if USE_ASYNC_LDS
// Builtin signature uses int* in AS1 (global) / AS3 (LDS).
#define ASYNC_CP_B32(lds_ptr, g_ptr)                                          \
  __builtin_amdgcn_global_load_async_to_lds_b32(                              \
      (__attribute__((address_space(1))) int*)                                \
          (__attribute__((address_space(1))) void*)(void*)(g_ptr),            \
      (__attribute__((address_space(3))) int*)                                \
          (__attribute__((address_space(3))) void*)(void*)(lds_ptr),          \
      0, 0)
static __device__ inline void wait_async_lds() {
#if __has_builtin(__builtin_amdgcn_s_wait_asynccnt)
  __builtin_amdgcn_s_wait_asynccnt(0);
#else
  asm volatile("s_wait_asynccnt 0x0" ::: "memory");
#endif
}
#endif

// ---------------------------------------------------------------- clamp copy
__global__ void clamp_copy_kernel(const float4* __restrict__ P,
                                  float4* __restrict__ LU) {
  int i = blockIdx.x * 256 + threadIdx.x;
  float4 v = P[i];
  v.x = fmaxf(v.x, EPSF); v.y = fmaxf(v.y, EPSF);
  v.z = fmaxf(v.z, EPSF); v.w = fmaxf(v.w, EPSF);
  LU[i] = v;
}

// ---------------------------------------------------------------- init vec
__global__ void init_vec_kernel(const float* __restrict__ innov,
                                float* __restrict__ b,
                                float* __restrict__ quad) {
  int i = blockIdx.x * 256 + threadIdx.x;
  if (i < Dn) b[i] = innov[i];
  if (i == 0) *quad = 0.0f;
}

// ------------------------------------------------- diagonal block LU (128x128)
__global__ void __launch_bounds__(256) lu_diag_kernel(float* __restrict__ LU, int kb) {
  __shared__ float s[NB * NB];                       // 64 KB
  const int tid = threadIdx.x;
  for (int e = tid; e < NB * NB; e += 256) {
    int r = e >> 7, c = e & 127;
    s[e] = LU[(long long)(kb + r) * Dn + kb + c];
  }
  __syncthreads();
  for (int k = 0; k < NB - 1; ++k) {
    float piv = s[k * NB + k];
    for (int i = k + 1 + tid; i < NB; i += 256) s[i * NB + k] /= piv;
    __syncthreads();
    int n = NB - 1 - k;
    for (int e = tid; e < n * n; e += 256) {
      int i = k + 1 + e / n, j = k + 1 + e % n;
      s[i * NB + j] -= s[i * NB + k] * s[k * NB + j];
    }
    __syncthreads();
  }
  for (int e = tid; e < NB * NB; e += 256) {
    int r = e >> 7, c = e & 127;
    LU[(long long)(kb + r) * Dn + kb + c] = s[e];
  }
}

// --------------------------------- U12 = L11^-1 * A12 (forward-subst columns)
#define TN 64
__global__ void __launch_bounds__(256) trsm_right_kernel(float* __restrict__ LU, int kb) {
  __shared__ float s[NB * TN];                       // 32 KB
  __shared__ float lc[NB];
  const int tid = threadIdx.x;
  long long col0 = (long long)(kb + NB) + (long long)blockIdx.x * TN;
  for (int e = tid; e < NB * TN; e += 256) {
    int r = e >> 6, c = e & 63;
    s[e] = LU[(long long)(kb + r) * Dn + col0 + c];
  }
  __syncthreads();
  for (int k = 0; k < NB - 1; ++k) {
    for (int j = k + 1 + tid; j < NB; j += 256)
      lc[j] = LU[(long long)(kb + j) * Dn + kb + k];
    __syncthreads();
    int n = NB - 1 - k;
    for (int e = tid; e < n * TN; e += 256) {
      int j = k + 1 + (e >> 6), c = e & 63;
      s[j * TN + c] -= lc[j] * s[k * TN + c];
    }
    __syncthreads();
  }
  for (int e = tid; e < NB * TN; e += 256) {
    int r = e >> 6, c = e & 63;
    LU[(long long)(kb + r) * Dn + col0 + c] = s[e];
  }
}

// ------------------------------------ L21 = A21 * U11^-1 (row-wise backsubst)
#define TM 64
__global__ void __launch_bounds__(256) trsm_left_kernel(float* __restrict__ LU, int kb) {
  __shared__ float s[TM * NB];                       // 32 KB
  __shared__ float ur[NB];
  const int tid = threadIdx.x;
  long long row0 = (long long)(kb + NB) + (long long)blockIdx.x * TM;
  for (int e = tid; e < TM * NB; e += 256) {
    int r = e >> 7, c = e & 127;
    s[e] = LU[(row0 + r) * Dn + kb + c];
  }
  __syncthreads();
  for (int k = 0; k < NB; ++k) {
    for (int c = tid; c < NB; c += 256)
      ur[c] = LU[(long long)(kb + k) * Dn + kb + c];
    __syncthreads();
    float inv = 1.0f / ur[k];
    for (int r = tid; r < TM; r += 256) s[r * NB + k] *= inv;
    __syncthreads();
    int n = NB - 1 - k;
    for (int e = tid; e < TM * n; e += 256) {
      int r = e / n, j = k + 1 + e % n;
      s[r * NB + j] -= s[r * NB + k] * ur[j];
    }
    __syncthreads();
  }
  for (int e = tid; e < TM * NB; e += 256) {
    int r = e >> 7, c = e & 127;
    LU[(row0 + r) * Dn + kb + c] = s[e];
  }
}

// ---------------------------- trailing update A22 -= L21*U12 via f32 WMMA
// Block tile 128x128, 8 waves, each wave 32x64 (2x4 accumulators).
// LDS tiles are k-pair-interleaved: element (k,m) lives at
// ((k>>1)*BM + m)*2 + (k&1), so a 16x4 fragment half for lanes 0-15 (K=k,k+1)
// / lanes 16-31 (K=k+2,k+3) is one aligned b64 load straight into a v2f.
#define BM 128
#define BN 128
#define BK 32
__global__ void __launch_bounds__(256) gemm_update_kernel(float* __restrict__ LU, int kb) {
  __shared__ __align__(16) float sa[2][BK * BM];     // 2 x 16 KB
  __shared__ __align__(16) float sb[2][BK * BN];     // 2 x 16 KB
  const int tid  = threadIdx.x;
  const int wave = tid >> 5;
  const int lane = tid & 31;
  const int l15  = lane & 15;
  const int ksel = lane >> 4;                        // 0: K=k,k+1  1: K=k+2,k+3
  const int wm   = (wave & 3) * 32;                  // 4 waves along M
  const int wn   = (wave >> 2) * 64;                 // 2 waves along N
  const long long m0g = (long long)(kb + NB) + (long long)blockIdx.x * BM;
  const long long n0g = (long long)(kb + NB) + (long long)blockIdx.y * BN;

  // Per-thread staging coordinates (coalesced 128B row segments).
  const int ka = tid & 31;           // A: fixed k, m = i*8 + tid>>5
  const int na = tid & 127;          // B: fixed n, k = i*2 + tid>>7

  v8f acc[2][4] = {};

  auto compute_chunk = [&](int c) {
#pragma unroll
    for (int kk = 0; kk < BK; kk += 4) {
      const int p = (kk >> 1) + ksel;
      v2f a0 = *(const v2f*)&sa[c][(p * BM + wm + l15) * 2];
      v2f a1 = *(const v2f*)&sa[c][(p * BM + wm + 16 + l15) * 2];
#pragma unroll
      for (int j = 0; j < 4; ++j) {
        v2f bj = *(const v2f*)&sb[c][(p * BN + wn + j * 16 + l15) * 2];
        acc[0][j] = __builtin_amdgcn_wmma_f32_16x16x4_f32(false, a0, false, bj,
                                                          (short)0, acc[0][j], false, false);
        acc[1][j] = __builtin_amdgcn_wmma_f32_16x16x4_f32(false, a1, false, bj,
                                                          (short)0, acc[1][j], false, false);
      }
    }
  };

#if USE_ASYNC_LDS
  // -------- CDNA5 async global->LDS pipeline (ASYNCcnt, no VGPR staging) ----
  auto issue_chunk = [&](int c, int k0) {
#pragma unroll
    for (int i = 0; i < 16; ++i) {
      int m = i * 8 + (tid >> 5);
      ASYNC_CP_B32(&sa[c][((ka >> 1) * BM + m) * 2 + (ka & 1)],
                   &LU[(m0g + m) * Dn + (kb + k0 + ka)]);
    }
#pragma unroll
    for (int i = 0; i < 16; ++i) {
      int k = i * 2 + (tid >> 7);
      ASYNC_CP_B32(&sb[c][((k >> 1) * BN + na) * 2 + (k & 1)],
                   &LU[(long long)(kb + k0 + k) * Dn + n0g + na]);
    }
  };
  issue_chunk(0, 0);
  int c = 0;
  for (int k0 = 0; k0 < NB; k0 += BK, c ^= 1) {
    wait_async_lds();
    __syncthreads();                 // copies visible; prev readers of c^1 done
    if (k0 + BK < NB) issue_chunk(c ^ 1, k0 + BK);
    compute_chunk(c);
  }
#else
  // -------- fallback: register-staged pipeline -----------------------------
  float ra[16], rb[16];
  auto load_chunk = [&](int k0) {
#pragma unroll
    for (int i = 0; i < 16; ++i) {
      int m = i * 8 + (tid >> 5);
      ra[i] = LU[(m0g + m) * Dn + (kb + k0 + ka)];
    }
#pragma unroll
    for (int i = 0; i < 16; ++i) {
      int k = i * 2 + (tid >> 7);
      rb[i] = LU[(long long)(kb + k0 + k) * Dn + n0g + na];
    }
  };
  auto store_chunk = [&](int c) {
#pragma unroll
    for (int i = 0; i < 16; ++i) {
      int m = i * 8 + (tid >> 5);
      sa[c][((ka >> 1) * BM + m) * 2 + (ka & 1)] = ra[i];
    }
#pragma unroll
    for (int i = 0; i < 16; ++i) {
      int k = i * 2 + (tid >> 7);
      sb[c][((k >> 1) * BN + na) * 2 + (k & 1)] = rb[i];
    }
  };
  load_chunk(0);
  int c = 0;
  for (int k0 = 0; k0 < NB; k0 += BK, c ^= 1) {
    store_chunk(c);
    __syncthreads();
    if (k0 + BK < NB) load_chunk(k0 + BK);   // overlap global latency with WMMA
    compute_chunk(c);
  }
#endif

  // C -= acc. C/D layout: VGPR r -> row r (lanes 0-15) or r+8 (lanes 16-31).
  const int rbase = ksel * 8;
#pragma unroll
  for (int mi = 0; mi < 2; ++mi)
#pragma unroll
    for (int j = 0; j < 4; ++j)
#pragma unroll
      for (int r = 0; r < 8; ++r) {
        long long row = m0g + wm + mi * 16 + rbase + r;
        long long col = n0g + wn + j * 16 + l15;
        LU[row * Dn + col] -= acc[mi][j][r];
      }
}

// ----------------------------------------------- forward solve: L y = v
__global__ void __launch_bounds__(128) fsolve_diag_kernel(const float* __restrict__ LU,
                                                          float* __restrict__ b, int kb) {
  __shared__ float y[NB];
  int t = threadIdx.x;
  y[t] = b[kb + t];
  __syncthreads();
  for (int k = 0; k < NB - 1; ++k) {
    if (t > k) y[t] -= LU[(long long)(kb + t) * Dn + kb + k] * y[k];
    __syncthreads();
  }
  b[kb + t] = y[t];
}

__global__ void __launch_bounds__(128) fsolve_update_kernel(const float* __restrict__ LU,
                                                            float* __restrict__ b, int kb) {
  __shared__ float ys[NB];
  int t = threadIdx.x;
  ys[t] = b[kb + t];
  __syncthreads();
  long long row = (long long)(kb + NB) + (long long)blockIdx.x * 128 + t;
  const float* Lrow = LU + row * Dn + kb;
  float sum = 0.0f;
#pragma unroll 8
  for (int k = 0; k < NB; ++k) sum += Lrow[k] * ys[k];
  b[row] -= sum;
}

// ----------------------------------------------- backward solve: U x = y
__global__ void __launch_bounds__(128) bsolve_diag_kernel(const float* __restrict__ LU,
                                                          float* __restrict__ b, int kb) {
  __shared__ float x[NB];
  int t = threadIdx.x;
  x[t] = b[kb + t];
  __syncthreads();
  for (int k = NB - 1; k >= 0; --k) {
    if (t == k) x[t] /= LU[(long long)(kb + k) * Dn + kb + k];
    __syncthreads();
    if (t < k) x[t] -= LU[(long long)(kb + t) * Dn + kb + k] * x[k];
    __syncthreads();
  }
  b[kb + t] = x[t];
}

__global__ void __launch_bounds__(128) bsolve_update_kernel(const float* __restrict__ LU,
                                                            float* __restrict__ b, int kb) {
  __shared__ float xs[NB];
  int t = threadIdx.x;
  xs[t] = b[kb + t];
  __syncthreads();
  long long row = (long long)blockIdx.x * 128 + t;   // rows [0, kb)
  const float* Urow = LU + row * Dn + kb;
  float sum = 0.0f;
#pragma unroll 8
  for (int k = 0; k < NB; ++k) sum += Urow[k] * xs[k];
  b[row] -= sum;
}

// ----------------------------------------------- quad = v . x (deterministic)
__global__ void __launch_bounds__(256) dot_kernel(const float* __restrict__ v,
                                                  const float* __restrict__ x,
                                                  float* __restrict__ quad) {
  __shared__ float red[256];
  int t = threadIdx.x;
  float s = 0.0f;
  for (int i = t; i < Dn; i += 256) s += v[i] * x[i];
  red[t] = s;
  __syncthreads();
  for (int o = 128; o > 0; o >>= 1) {
    if (t < o) red[t] += red[t + o];
    __syncthreads();
  }
  if (t == 0) quad[0] = red[0];
}

// --------------------- loss = 0.5*log(2*pi*Pc) + 0.5*quad (rewrites all d_out)
__global__ void final_kernel(const float4* __restrict__ P,
                             float4* __restrict__ out,
                             const float* __restrict__ quad) {
  float q = 0.5f * quad[0];
  int i = blockIdx.x * 256 + threadIdx.x;
  float4 p = P[i];
  float4 o;
  o.x = 0.5f * logf(TWOPI_F * fmaxf(p.x, EPSF)) + q;
  o.y = 0.5f * logf(TWOPI_F * fmaxf(p.y, EPSF)) + q;
  o.z = 0.5f * logf(TWOPI_F * fmaxf(p.z, EPSF)) + q;
  o.w = 0.5f * logf(TWOPI_F * fmaxf(p.w, EPSF)) + q;
  out[i] = o;
}

extern "C" void kernel_launch(void* const* d_in, const int* in_sizes, int n_in,
                              void* d_out, int out_size, void* d_ws, size_t ws_size,
                              hipStream_t stream) {
  (void)in_sizes; (void)n_in; (void)out_size; (void)ws_size;
  const float* innov = (const float*)d_in[0];   // (D,1) fp32
  const float* P     = (const float*)d_in[1];   // (D,D) fp32
  float* out = (float*)d_out;
  float* LU  = out;                             // reuse output buffer as LU scratch
  float* b   = (float*)d_ws;                    // D floats: v -> y -> x in place
  float* quad = b + Dn;                         // 1 float

  const int elem4 = (Dn * Dn) / 4;

  clamp_copy_kernel<<<elem4 / 256, 256, 0, stream>>>((const float4*)P, (float4*)LU);
  init_vec_kernel<<<Dn / 256, 256, 0, stream>>>(innov, b, quad);

  // Blocked right-looking LU (no pivoting), 32 panels of NB=128.
  for (int p = 0; p < Dn / NB; ++p) {
    int kb  = p * NB;
    int rem = Dn - kb - NB;
    lu_diag_kernel<<<1, 256, 0, stream>>>(LU, kb);
    if (rem > 0) {
      trsm_right_kernel<<<rem / TN, 256, 0, stream>>>(LU, kb);
      trsm_left_kernel<<<rem / TM, 256, 0, stream>>>(LU, kb);
      gemm_update_kernel<<<dim3(rem / BM, rem / BN), 256, 0, stream>>>(LU, kb);
    }
  }

  // Forward solve L y = v.
  for (int p = 0; p < Dn / NB; ++p) {
    int kb  = p * NB;
    int rem = Dn - kb - NB;
    fsolve_diag_kernel<<<1, 128, 0, stream>>>(LU, b, kb);
    if (rem > 0) fsolve_update_kernel<<<rem / 128, 128, 0, stream>>>(LU, b, kb);
  }

  // Backward solve U x = y.
  for (int p = Dn / NB - 1; p >= 0; --p) {
    int kb = p * NB;
    bsolve_diag_kernel<<<1, 128, 0, stream>>>(LU, b, kb);
    if (kb > 0) bsolve_update_kernel<<<kb / 128, 128, 0, stream>>>(LU, b, kb);
  }

  dot_kernel<<<1, 256, 0, stream>>>(innov, b, quad);
  final_kernel<<<elem4 / 256, 256, 0, stream>>>((const float4*)P, (float4*)out, quad);
}